// AdaptiveLocallyDirected1D_22771916603540
// MI455X (gfx1250) — compile-verified
//
#include <hip/hip_runtime.h>

#define B_DIM 64
#define N_IN  1000000
#define N_OUT 20000
#define KMAX  64

typedef __attribute__((ext_vector_type(2))) float v2f;
typedef __attribute__((ext_vector_type(8))) float v8f;

// ---------------------------------------------------------------------------
// Pass 1: transpose x (B=64, N_IN) row-major -> xT (N_IN, 64) row-major.
// Fully vectorized: b128 global loads, transpose into LDS, b128 global stores.
// Each gathered "column" of x becomes one contiguous 256B row of xT.
// ---------------------------------------------------------------------------
__global__ __launch_bounds__(256)
void transpose64_kernel(const float* __restrict__ x, float* __restrict__ xT) {
    __shared__ float tileT[64][65];          // tileT[j][b]; pad -> conflict-free
    const int j0 = blockIdx.x * 64;          // N_IN = 15625 * 64, exact
    const int t  = threadIdx.x;
#pragma unroll
    for (int i = 0; i < 4; ++i) {
        const int q  = t + 256 * i;          // 0..1023
        const int r  = q >> 4;               // batch row b, 0..63
        const int cq = q & 15;               // float4 slot within row
        const float4 v = *reinterpret_cast<const float4*>(
            x + (size_t)r * N_IN + (j0 + cq * 4));
        tileT[cq * 4 + 0][r] = v.x;          // transpose on the way into LDS
        tileT[cq * 4 + 1][r] = v.y;
        tileT[cq * 4 + 2][r] = v.z;
        tileT[cq * 4 + 3][r] = v.w;
    }
    __syncthreads();
#pragma unroll
    for (int i = 0; i < 4; ++i) {
        const int q  = t + 256 * i;
        const int jj = q >> 4;               // xT row, 0..63
        const int cb = q & 15;               // float4 slot within row
        float4 v;
        v.x = tileT[jj][cb * 4 + 0];
        v.y = tileT[jj][cb * 4 + 1];
        v.z = tileT[jj][cb * 4 + 2];
        v.w = tileT[jj][cb * 4 + 3];
        *reinterpret_cast<float4*>(xT + (size_t)(j0 + jj) * 64 + cb * 4) = v;
    }
}

// ---------------------------------------------------------------------------
// Pass 2: one wave32 per output g.
//   out[b,g] = sum_k xsrc[idx[g,k]*jstride + b*bstride] * w[g,k] + bias[g]
// Reduction via V_WMMA_F32_16X16X4_F32:
//   A (16x4)  = gathered tile G[b,k] (ISA layout: lanes 0-15 K=0,1;
//                                      lanes 16-31 K=2,3)
//   B (4x16)  = w[k] replicated across N (every column of D = full dot)
//   C/D accumulated across 16 K-tiles; 4 batch tiles of 16.
// All 64 gather rows are prefetched into L2 before the compute loop.
// ---------------------------------------------------------------------------
__global__ __launch_bounds__(256)
void gather_mv_wmma_kernel(const float* __restrict__ xsrc,
                           const float* __restrict__ kern,
                           const float* __restrict__ bias,
                           const int*   __restrict__ gidx,
                           const int*   __restrict__ lens,
                           float*       __restrict__ out,
                           long jstride, long bstride) {
    __shared__ int   sIdx[8][KMAX];
    __shared__ float sW[8][KMAX];

    const int wave = threadIdx.x >> 5;
    const int lane = threadIdx.x & 31;
    const int g    = blockIdx.x * 8 + wave;          // N_OUT = 2500 * 8, exact

    // Stage per-g indices and masked weights (uniform across the wave).
    const int len = lens[g];
    const int kA = lane, kB = lane + 32;
    const int ia = gidx[(size_t)g * KMAX + kA];
    const int ib = gidx[(size_t)g * KMAX + kB];
    sIdx[wave][kA] = ia;
    sIdx[wave][kB] = ib;

    // Prefetch the full 16KB gather footprint (64 rows x 256B) into L2:
    // 4 prefetch instructions cover it (2 rows/lane x 2 cachelines/row).
    const char* pa = (const char*)(xsrc + (long)ia * jstride);
    const char* pb = (const char*)(xsrc + (long)ib * jstride);
    __builtin_prefetch(pa, 0, 3);
    __builtin_prefetch(pa + 128, 0, 3);
    __builtin_prefetch(pb, 0, 3);
    __builtin_prefetch(pb + 128, 0, 3);

    sW[wave][kA] = (kA < len) ? kern[(size_t)kA * N_OUT + g] : 0.0f;
    sW[wave][kB] = (kB < len) ? kern[(size_t)kB * N_OUT + g] : 0.0f;
    __syncthreads();

    const int half = lane >> 4;          // 0: K pair {0,1}, 1: K pair {2,3}
    const int sub  = lane & 15;          // A-matrix row within 16-batch tile

    v8f acc[4];
#pragma unroll
    for (int bt = 0; bt < 4; ++bt) acc[bt] = (v8f){0,0,0,0,0,0,0,0};

#pragma unroll 8
    for (int t = 0; t < 16; ++t) {       // 16 K-tiles of 4
        const int ka = 4 * t + 2 * half;
        const long ja = (long)sIdx[wave][ka]     * jstride;   // LDS broadcast
        const long jb = (long)sIdx[wave][ka + 1] * jstride;
        v2f Bv;
        Bv.x = sW[wave][ka];             // B[k][n] = w[k] for all n
        Bv.y = sW[wave][ka + 1];
#pragma unroll
        for (int bt = 0; bt < 4; ++bt) {
            const long col = (long)(bt * 16 + sub) * bstride;
            v2f Av;
            Av.x = xsrc[ja + col];       // G[b, ka]
            Av.y = xsrc[jb + col];       // G[b, ka+1]
            acc[bt] = __builtin_amdgcn_wmma_f32_16x16x4_f32(
                false, Av, false, Bv, (short)0, acc[bt], false, false);
        }
    }

    // Every column of D holds the full dot product. C/D layout: VGPR r holds
    // M=r (lanes 0-15) and M=8+r (lanes 16-31). Lanes with sub==0 write out.
    const float bg = bias[g];
    if (sub == 0) {
        const int mb = half * 8;
#pragma unroll
        for (int r = 0; r < 8; ++r) {
#pragma unroll
            for (int bt = 0; bt < 4; ++bt) {
                out[(size_t)(bt * 16 + mb + r) * N_OUT + g] = acc[bt][r] + bg;
            }
        }
    }
}

// ---------------------------------------------------------------------------
extern "C" void kernel_launch(void* const* d_in, const int* in_sizes, int n_in,
                              void* d_out, int out_size, void* d_ws, size_t ws_size,
                              hipStream_t stream) {
    (void)in_sizes; (void)n_in; (void)out_size;
    const float* x    = (const float*)d_in[0];   // (64, 1e6) f32
    const float* kern = (const float*)d_in[1];   // (64, 20000) f32
    const float* bias = (const float*)d_in[2];   // (20000, 1) f32
    const int*   gidx = (const int*)d_in[3];     // (20000, 64) i32
    const int*   lens = (const int*)d_in[4];     // (20000,) i32
    float*       out  = (float*)d_out;           // (64, 20000, 1) f32

    const size_t xT_bytes = (size_t)N_IN * B_DIM * sizeof(float); // 256 MB
    if (ws_size >= xT_bytes) {
        float* xT = (float*)d_ws;
        transpose64_kernel<<<N_IN / 64, 256, 0, stream>>>(x, xT);
        gather_mv_wmma_kernel<<<N_OUT / 8, 256, 0, stream>>>(
            xT, kern, bias, gidx, lens, out, /*jstride=*/B_DIM, /*bstride=*/1);
    } else {
        // Fallback: gather directly from x (strided, slower, still correct).
        gather_mv_wmma_kernel<<<N_OUT / 8, 256, 0, stream>>>(
            x, kern, bias, gidx, lens, out, /*jstride=*/1, /*bstride=*/N_IN);
    }
}